// PointnetFPModule_1082331758732
// MI455X (gfx1250) — compile-verified
//
#include <hip/hip_runtime.h>

// Problem constants (from setup_inputs)
#define BB    16
#define NPTS  4096
#define MPTS  1024
#define C1    256
#define C2    256
#define HCH   256
#define NCOLS (BB * NPTS)   // 65536
#define K1    (C1 + C2)     // 512

typedef __attribute__((ext_vector_type(2))) float v2f;
typedef __attribute__((ext_vector_type(8))) float v8f;

// ---------------------------------------------------------------------------
// CDNA5 async global->LDS copy (tracked by ASYNCcnt), 16B per lane.
// LDS destination address = low 32 bits of the generic pointer to __shared__
// (generic LDS pointers are {SHARED_BASE, offset} on AMDGPU).
// ---------------------------------------------------------------------------
__device__ __forceinline__ void async_b128(void* lds, const void* gaddr) {
  const unsigned l = (unsigned)(unsigned long long)lds;
  asm volatile("global_load_async_to_lds_b128 %0, %1, off"
               :: "v"(l), "v"(gaddr) : "memory");
}
__device__ __forceinline__ void wait_async_le4() {
  asm volatile("s_wait_asynccnt 0x4" ::: "memory");
}
__device__ __forceinline__ void wait_async_0() {
  asm volatile("s_wait_asynccnt 0x0" ::: "memory");
}

// ---------------------------------------------------------------------------
// Kernel 1: three_nn + weights. One block = 256 unknown points of one batch.
// Known points staged in LDS (SoA, 12KB). Stable top-3 (matches lax.top_k ties).
// ---------------------------------------------------------------------------
__global__ __launch_bounds__(256) void three_nn_kernel(
    const float* __restrict__ unknown, const float* __restrict__ known,
    int* __restrict__ idx, float* __restrict__ wgt) {
  __shared__ float kx[MPTS], ky[MPTS], kz[MPTS];
  const int b = blockIdx.y;
  const int t = threadIdx.x;
  const float* kb = known + b * MPTS * 3;
  for (int p = t; p < MPTS; p += 256) {
    kx[p] = kb[p * 3 + 0];
    ky[p] = kb[p * 3 + 1];
    kz[p] = kb[p * 3 + 2];
  }
  __syncthreads();

  const int p = blockIdx.x * 256 + t;
  const float* up = unknown + (b * NPTS + p) * 3;
  const float ux = up[0], uy = up[1], uz = up[2];

  float d0 = 3.4e38f, d1v = 3.4e38f, d2v = 3.4e38f;
  int i0 = 0, i1 = 0, i2 = 0;
  for (int q = 0; q < MPTS; ++q) {
    const float dx = ux - kx[q], dy = uy - ky[q], dz = uz - kz[q];
    const float d = dx * dx + dy * dy + dz * dz;
    if (d < d0)       { d2v = d1v; i2 = i1; d1v = d0; i1 = i0; d0 = d;  i0 = q; }
    else if (d < d1v) { d2v = d1v; i2 = i1; d1v = d;  i1 = q; }
    else if (d < d2v) { d2v = d;  i2 = q; }
  }
  const float r0 = 1.0f / (d0 + 1e-8f);
  const float r1 = 1.0f / (d1v + 1e-8f);
  const float r2 = 1.0f / (d2v + 1e-8f);
  const float inv = 1.0f / (r0 + r1 + r2);
  const int base = (b * NPTS + p) * 3;
  idx[base + 0] = i0; idx[base + 1] = i1; idx[base + 2] = i2;
  wgt[base + 0] = r0 * inv; wgt[base + 1] = r1 * inv; wgt[base + 2] = r2 * inv;
}

// ---------------------------------------------------------------------------
// Kernel 2: three_interpolate + concat -> X (row-major [512][65536]).
// Rows 0..255 = interpolated known_feats, rows 256..511 = unknow_feats.
// tid == c*65536 + col so X[tid] is the natural store.
// ---------------------------------------------------------------------------
__global__ __launch_bounds__(256) void interp_concat_kernel(
    const float* __restrict__ unknow_feats, const float* __restrict__ known_feats,
    const int* __restrict__ idx, const float* __restrict__ wgt,
    float* __restrict__ X) {
  const int tid = blockIdx.x * 256 + threadIdx.x;
  const int c = tid >> 16;            // NCOLS == 65536
  const int col = tid & (NCOLS - 1);
  const int b = col >> 12;            // NPTS == 4096
  const int j = col & (NPTS - 1);
  float v;
  if (c < C2) {
    const int base = (b * NPTS + j) * 3;
    const int i0 = idx[base], i1 = idx[base + 1], i2 = idx[base + 2];
    const float w0 = wgt[base], w1 = wgt[base + 1], w2 = wgt[base + 2];
    const float* f = known_feats + (b * C2 + c) * MPTS;
    v = w0 * f[i0] + w1 * f[i1] + w2 * f[i2];
  } else {
    v = unknow_feats[(b * C1 + (c - C2)) * NPTS + j];
  }
  X[tid] = v;
}

// ---------------------------------------------------------------------------
// Kernel 3: fp32 WMMA GEMM  Y(M x 65536) = W(M x K) * Xeff(K x 65536)
// Block tile 128x128, 8 waves in a 2(M) x 4(N) grid, each wave 64x32 =
// 4x2 tiles of V_WMMA_F32_16X16X4_F32.
// Tiles are double-buffered in LDS and staged with GLOBAL_LOAD_ASYNC_TO_LDS
// (ASYNCcnt): issue next chunk's 4 B128s/thread, s_wait_asynccnt 4 => the
// current chunk (issued earlier, in-order) is complete, barrier, compute.
// Optional fused BN+ReLU on the X operand is applied at fragment-read time
// (stage 2 consumes raw Y1 with stage-1 folded scale/bias from LDS).
// ---------------------------------------------------------------------------
#define BM 128
#define BNT 128
#define KB 16
#define AS_STRIDE 20    // 20 mod 64: conflict-free A fragment reads, 16B-aligned f4 slots
#define BS_STRIDE 136   // 136 mod 64 = 8: rows k and k+2 land in disjoint bank ranges

__global__ __launch_bounds__(256) void gemm_wmma_kernel(
    const float* __restrict__ W, const float* __restrict__ X,
    float* __restrict__ Y, int K,
    const float* __restrict__ in_scale, const float* __restrict__ in_bias) {
  __shared__ __align__(16) float As[2][BM * AS_STRIDE];
  __shared__ __align__(16) float Bs[2][KB * BS_STRIDE];
  __shared__ float sScale[K1];
  __shared__ float sBias[K1];

  const int t = threadIdx.x;
  const int lane = t & 31;
  const int w = t >> 5;
  const int wm = w & 1;        // 2 wave rows (64 each)
  const int wn = w >> 1;       // 4 wave cols (32 each)
  const int half = lane >> 4;  // selects K pair (A) / M offset (C)
  const int lid = lane & 15;
  const int mBase = blockIdx.y * BM;
  const int nBase = blockIdx.x * BNT;

  if (in_scale) {
    for (int i = t; i < K; i += 256) {
      sScale[i] = in_scale[i];
      sBias[i] = in_bias[i];
    }
  }

  // Per-thread async-copy slots: A tile 128x16 f32 = 512 float4 (2/thread),
  // B tile 16x128 f32 = 512 float4 (2/thread).
  const int rA = t >> 2;        // A row 0..63 (s=1 adds 64)
  const int cvA = t & 3;        // float4 within 16-float A row
  const int krB = t >> 5;       // B k-row 0..7 (s=1 adds 8)
  const int nvB = t & 31;       // float4 within 128-float B row
  const float* gA = W + (mBase + rA) * K + cvA * 4;          // + kt
  const float* gB = X + krB * NCOLS + nBase + nvB * 4;       // + kt*NCOLS

  v8f acc[4][2];
#pragma unroll
  for (int mi = 0; mi < 4; ++mi)
#pragma unroll
    for (int ni = 0; ni < 2; ++ni) {
      v8f z = {};
      acc[mi][ni] = z;
    }

  auto issue_tile = [&](int buf, int kt) {
    async_b128(&As[buf][rA * AS_STRIDE + cvA * 4],        gA + kt);
    async_b128(&As[buf][(rA + 64) * AS_STRIDE + cvA * 4], gA + 64 * K + kt);
    async_b128(&Bs[buf][krB * BS_STRIDE + nvB * 4],       gB + kt * NCOLS);
    async_b128(&Bs[buf][(krB + 8) * BS_STRIDE + nvB * 4], gB + (kt + 8) * NCOLS);
  };

  const int nChunks = K / KB;
  issue_tile(0, 0);

  for (int ic = 0; ic < nChunks; ++ic) {
    const int kt = ic * KB;
    const int cur = ic & 1;
    const bool more = (ic + 1) < nChunks;
    if (more) {
      issue_tile(cur ^ 1, kt + KB);
      wait_async_le4();          // 4 outstanding = next tile; current is done
    } else {
      wait_async_0();
    }
    __syncthreads();             // all waves' current-buffer loads visible

    const float* Ac = As[cur];
    const float* Bc = Bs[cur];
#pragma unroll
    for (int kk = 0; kk < KB; kk += 4) {
      const int kl = kk + 2 * half;  // lanes 0-15: K=kk,kk+1 ; 16-31: kk+2,kk+3
      v2f a[4];
#pragma unroll
      for (int mi = 0; mi < 4; ++mi) {
        const int mr = wm * 64 + mi * 16 + lid;
        a[mi].x = Ac[mr * AS_STRIDE + kl];
        a[mi].y = Ac[mr * AS_STRIDE + kl + 1];
      }
      v2f bf[2];
#pragma unroll
      for (int ni = 0; ni < 2; ++ni) {
        const int nc = wn * 32 + ni * 16 + lid;
        bf[ni].x = Bc[kl * BS_STRIDE + nc];
        bf[ni].y = Bc[(kl + 1) * BS_STRIDE + nc];
      }
      if (in_scale) {            // stage-2: H = relu(bn1(Y1)) on the fly
        const float s0 = sScale[kt + kl], b0 = sBias[kt + kl];
        const float s1 = sScale[kt + kl + 1], b1 = sBias[kt + kl + 1];
#pragma unroll
        for (int ni = 0; ni < 2; ++ni) {
          bf[ni].x = fmaxf(0.0f, bf[ni].x * s0 + b0);
          bf[ni].y = fmaxf(0.0f, bf[ni].y * s1 + b1);
        }
      }
#pragma unroll
      for (int mi = 0; mi < 4; ++mi)
#pragma unroll
        for (int ni = 0; ni < 2; ++ni)
          acc[mi][ni] = __builtin_amdgcn_wmma_f32_16x16x4_f32(
              false, a[mi], false, bf[ni], (short)0, acc[mi][ni], false, false);
    }
    __syncthreads();             // readers done before buffer is re-targeted
  }

  // Epilogue: C/D layout — VGPR r holds M=r (lanes 0-15) / M=8+r (lanes 16-31), N=lid.
#pragma unroll
  for (int mi = 0; mi < 4; ++mi) {
#pragma unroll
    for (int ni = 0; ni < 2; ++ni) {
      const int nc = nBase + wn * 32 + ni * 16 + lid;
      const int mb = mBase + wm * 64 + mi * 16 + half * 8;
#pragma unroll
      for (int r = 0; r < 8; ++r)
        Y[(mb + r) * NCOLS + nc] = acc[mi][ni][r];
    }
  }
}

// ---------------------------------------------------------------------------
// Kernel 4: per-channel BN stats -> folded scale/bias. One block per channel.
// mean over B*n (population variance, eps = 1e-5, as jnp).
// ---------------------------------------------------------------------------
__global__ __launch_bounds__(256) void bn_stats_kernel(
    const float* __restrict__ Y, const float* __restrict__ g,
    const float* __restrict__ be, float* __restrict__ scale,
    float* __restrict__ bias) {
  __shared__ float ss[256];
  __shared__ float ssq[256];
  const int c = blockIdx.x;
  const float* row = Y + (size_t)c * NCOLS;
  float s = 0.0f, sq = 0.0f;
  for (int i = threadIdx.x; i < NCOLS; i += 256) {
    const float v = row[i];
    s += v;
    sq += v * v;
  }
  ss[threadIdx.x] = s;
  ssq[threadIdx.x] = sq;
  __syncthreads();
  for (int off = 128; off > 0; off >>= 1) {
    if (threadIdx.x < off) {
      ss[threadIdx.x] += ss[threadIdx.x + off];
      ssq[threadIdx.x] += ssq[threadIdx.x + off];
    }
    __syncthreads();
  }
  if (threadIdx.x == 0) {
    const float mean = ss[0] * (1.0f / NCOLS);
    const float var = ssq[0] * (1.0f / NCOLS) - mean * mean;
    const float sc = g[c] * rsqrtf(var + 1e-5f);
    scale[c] = sc;
    bias[c] = be[c] - mean * sc;
  }
}

// ---------------------------------------------------------------------------
// Kernel 5: final BN+ReLU, remap columns (col = b*4096 + j) to (B, C, n) out.
// ---------------------------------------------------------------------------
__global__ __launch_bounds__(256) void bn_relu_out_kernel(
    const float* __restrict__ Y, const float* __restrict__ scale,
    const float* __restrict__ bias, float* __restrict__ out) {
  const int tid = blockIdx.x * 256 + threadIdx.x;
  const int c = tid >> 16;
  const int col = tid & (NCOLS - 1);
  const int b = col >> 12;
  const int j = col & (NPTS - 1);
  const float v = fmaxf(0.0f, Y[tid] * scale[c] + bias[c]);
  out[(b * HCH + c) * NPTS + j] = v;
}

// ---------------------------------------------------------------------------
// Launch
// ---------------------------------------------------------------------------
extern "C" void kernel_launch(void* const* d_in, const int* in_sizes, int n_in,
                              void* d_out, int out_size, void* d_ws, size_t ws_size,
                              hipStream_t stream) {
  const float* unknown      = (const float*)d_in[0];
  const float* known        = (const float*)d_in[1];
  const float* unknow_feats = (const float*)d_in[2];
  const float* known_feats  = (const float*)d_in[3];
  const float* W1 = (const float*)d_in[4];
  const float* g1 = (const float*)d_in[5];
  const float* be1 = (const float*)d_in[6];
  const float* W2 = (const float*)d_in[7];
  const float* g2 = (const float*)d_in[8];
  const float* be2 = (const float*)d_in[9];

  char* ws = (char*)d_ws;
  // Workspace layout (bytes):
  //   [0, 786432)            idx   : int  [B*n*3]
  //   [786432, 1572864)      wgt   : f32  [B*n*3]
  //   [1572864, +4KB)        folded BN scale/bias (4 x 256 f32)
  //   [2097152, +128MB)      X     : f32  [512][65536]   (later aliased by Y2)
  //   [2097152+128MB, +64MB) Y1    : f32  [256][65536]
  int*   idx    = (int*)ws;
  float* wgt    = (float*)(ws + 786432);
  float* scale1 = (float*)(ws + 1572864);
  float* bias1  = scale1 + 256;
  float* scale2 = bias1 + 256;
  float* bias2  = scale2 + 256;
  float* X  = (float*)(ws + 2097152);
  float* Y1 = (float*)(ws + 2097152 + (size_t)K1 * NCOLS * 4);
  float* Y2 = X;  // X is dead after GEMM1; reuse for Y2

  three_nn_kernel<<<dim3(NPTS / 256, BB), 256, 0, stream>>>(unknown, known, idx, wgt);

  interp_concat_kernel<<<(K1 * NCOLS) / 256, 256, 0, stream>>>(
      unknow_feats, known_feats, idx, wgt, X);

  gemm_wmma_kernel<<<dim3(NCOLS / BNT, HCH / BM), 256, 0, stream>>>(
      W1, X, Y1, K1, nullptr, nullptr);

  bn_stats_kernel<<<HCH, 256, 0, stream>>>(Y1, g1, be1, scale1, bias1);

  gemm_wmma_kernel<<<dim3(NCOLS / BNT, HCH / BM), 256, 0, stream>>>(
      W2, Y1, Y2, HCH, scale1, bias1);

  bn_stats_kernel<<<HCH, 256, 0, stream>>>(Y2, g2, be2, scale2, bias2);

  bn_relu_out_kernel<<<(HCH * NCOLS) / 256, 256, 0, stream>>>(
      Y2, scale2, bias2, (float*)d_out);
}